// MYNET_51780125720793
// MI455X (gfx1250) — compile-verified
//
#include <hip/hip_runtime.h>
#include <math.h>

typedef float v2f __attribute__((ext_vector_type(2)));
typedef float v8f __attribute__((ext_vector_type(8)));
typedef int   v4i __attribute__((ext_vector_type(4)));

#define NCLS  100
#define DFEAT 512
#define RESO  25
#define BATCH 128
#define NQ    (BATCH * RESO)   // 3200

// ---- optional CDNA5 async global->LDS staging (guarded; sync fallback) ----
#if defined(__has_builtin)
#if __has_builtin(__builtin_amdgcn_global_load_async_to_lds_b128)
#define HAVE_ASYNC_LDS 1
#endif
#endif

#ifdef HAVE_ASYNC_LDS
typedef __attribute__((address_space(1))) v4i v4i_glb;   // clang prints as "__device__"
typedef __attribute__((address_space(3))) v4i v4i_lds;   // clang prints as "__shared__"
__device__ __forceinline__ void async_copy16(const void* g, void* l) {
  __builtin_amdgcn_global_load_async_to_lds_b128((v4i_glb*)g, (v4i_lds*)l, 0, 0);
}
__device__ __forceinline__ void wait_async0() {
#if __has_builtin(__builtin_amdgcn_s_wait_asynccnt)
  __builtin_amdgcn_s_wait_asynccnt(0);
#else
  asm volatile("s_wait_asynccnt 0x0" ::: "memory");
#endif
}
#endif

// ---------------- kernel: zero accumulator ----------------
__global__ void k_zero(float* __restrict__ p, int n) {
  int i = blockIdx.x * blockDim.x + threadIdx.x;
  if (i < n) p[i] = 0.0f;
}

// ---------------- kernel: build query [N,d] + qnorm [N] ----------------
__global__ void k_query(const float* __restrict__ x,
                        float* __restrict__ query,
                        float* __restrict__ qnorm) {
  const int n = blockIdx.x;            // 0..3199
  const int b = n / RESO, s = n % RESO;
  const float isq = 0.044194173824159216f;   // 1/sqrt(512)
  __shared__ float red[128];
  float sum = 0.0f;
  for (int j = threadIdx.x; j < DFEAT; j += 128) {
    float v = x[(b * DFEAT + j) * RESO + s] * isq;
    query[n * DFEAT + j] = v;
    sum += v * v;
  }
  red[threadIdx.x] = sum;
  __syncthreads();
  for (int off = 64; off > 0; off >>= 1) {
    if (threadIdx.x < off) red[threadIdx.x] += red[threadIdx.x + off];
    __syncthreads();
  }
  if (threadIdx.x == 0) qnorm[n] = red[0];
}

// ---------------- kernel: per-class (S S^T + lam I)^-1, zero-padded 32x32 ----------------
__global__ void k_inv(const float* __restrict__ proto,
                      const float* __restrict__ r,
                      float* __restrict__ invP) {
  const int c = blockIdx.x;
  const int t = threadIdx.x;           // 256 threads
  __shared__ float M[RESO][52];        // augmented [A | I]
  __shared__ float fcol[RESO];
  const float lam = __expf(r[0]) * ((float)RESO / (float)DFEAT) + 1e-6f;
  const float* S = proto + (size_t)c * RESO * DFEAT;

  for (int e = t; e < RESO * RESO; e += 256) {
    int i = e / RESO, j = e % RESO;
    float dot = 0.0f;
    for (int k = 0; k < DFEAT; ++k) dot += S[i * DFEAT + k] * S[j * DFEAT + k];
    M[i][j] = dot + (i == j ? lam : 0.0f);
    M[i][RESO + j] = (i == j) ? 1.0f : 0.0f;
  }
  __syncthreads();

  // Gauss-Jordan, no pivoting (SPD + lam I)
  for (int k = 0; k < RESO; ++k) {
    float pinv = 1.0f / M[k][k];
    __syncthreads();
    if (t < 2 * RESO) M[k][t] *= pinv;
    if (t < RESO) fcol[t] = (t == k) ? 0.0f : M[t][k];
    __syncthreads();
    for (int e = t; e < RESO * 2 * RESO; e += 256) {
      int i = e / (2 * RESO), j = e % (2 * RESO);
      if (i != k) M[i][j] -= fcol[i] * M[k][j];
    }
    __syncthreads();
  }

  for (int e = t; e < 32 * 32; e += 256) {
    int i = e >> 5, j = e & 31;
    invP[(size_t)c * 1024 + e] = (i < RESO && j < RESO) ? M[i][RESO + j] : 0.0f;
  }
}

// ---------------- main WMMA kernel: one class x 128 query rows per block ----------------
// Wave tile: 32 rows x 32 cols (2x2 blocking of 16x16x4 f32 WMMA).
__global__ void __launch_bounds__(128)
k_main(const float* __restrict__ query, const float* __restrict__ proto,
       const float* __restrict__ r, const float* __restrict__ invP,
       float* __restrict__ accOut) {
  const int c     = blockIdx.y;
  const int n0    = blockIdx.x * 128;
  const int t     = threadIdx.x;
  const int lane  = t & 31;
  const int wid   = t >> 5;
  const int half  = lane >> 4;          // K-subgroup within fragment
  const int col   = lane & 15;          // B/C/D column (and A-row index)
  const int rbase = wid * 32;           // wave's 32-row tile base
  const int mrow0 = rbase + col;        // A fragment 0 row
  const int mrow1 = mrow0 + 16;         // A fragment 1 row

  // LDS: sQ (128x36) aliased with sU; sS (32x36) aliased with sInv; sW (128x36)
  __shared__ float smem[128 * 36 + 32 * 36 + 128 * 36];   // 41472 B
  float (*sQ)[36]   = reinterpret_cast<float(*)[36]>(smem);
  float (*sU)[36]   = reinterpret_cast<float(*)[36]>(smem);
  float (*sS)[36]   = reinterpret_cast<float(*)[36]>(smem + 128 * 36);
  float (*sInv)[36] = reinterpret_cast<float(*)[36]>(smem + 128 * 36);
  float (*sW)[36]   = reinterpret_cast<float(*)[36]>(smem + 128 * 36 + 32 * 36);

  v8f a00 = {}, a01 = {}, a10 = {}, a11 = {};   // U accumulators
  const float* Sc = proto + (size_t)c * RESO * DFEAT;

#ifdef HAVE_ASYNC_LDS
  // zero the pad rows of sS once (never overwritten by async loads below)
  for (int idx = t; idx < 7 * 32; idx += 128)
    sS[RESO + (idx >> 5)][idx & 31] = 0.0f;
#endif

  // ---- Phase 1: U = Q_tile(128x512) @ S_c^T(512x32pad), K chunked by 32 ----
  for (int k0 = 0; k0 < DFEAT; k0 += 32) {
#ifdef HAVE_ASYNC_LDS
    for (int idx = t; idx < 1024; idx += 128) {          // 128x32 floats, 16B/lane
      int rr = idx >> 3, k4 = (idx & 7) << 2;
      async_copy16(&query[(size_t)(n0 + rr) * DFEAT + k0 + k4], &sQ[rr][k4]);
    }
    for (int idx = t; idx < 200; idx += 128) {           // 25x32 floats, 16B/lane
      int rc = idx >> 3, k4 = (idx & 7) << 2;
      async_copy16(&Sc[rc * DFEAT + k0 + k4], &sS[rc][k4]);
    }
    wait_async0();
#else
    for (int idx = t; idx < 1024; idx += 128) {          // 128x32 floats as float4
      int rr = idx >> 3, k4 = (idx & 7) << 2;
      *reinterpret_cast<float4*>(&sQ[rr][k4]) =
          *reinterpret_cast<const float4*>(&query[(size_t)(n0 + rr) * DFEAT + k0 + k4]);
    }
    for (int idx = t; idx < 256; idx += 128) {           // 32x32 floats as float4
      int rc = idx >> 3, k4 = (idx & 7) << 2;
      float4 v = make_float4(0.f, 0.f, 0.f, 0.f);
      if (rc < RESO) v = *reinterpret_cast<const float4*>(&Sc[rc * DFEAT + k0 + k4]);
      *reinterpret_cast<float4*>(&sS[rc][k4]) = v;       // natural layout: sS[protoRow][k]
    }
#endif
    __syncthreads();
    for (int kk = 0; kk < 32; kk += 4) {
      v2f a0 = *reinterpret_cast<const v2f*>(&sQ[mrow0][kk + 2 * half]);
      v2f a1 = *reinterpret_cast<const v2f*>(&sQ[mrow1][kk + 2 * half]);
      v2f b0 = *reinterpret_cast<const v2f*>(&sS[col][kk + 2 * half]);
      v2f b1 = *reinterpret_cast<const v2f*>(&sS[col + 16][kk + 2 * half]);
      a00 = __builtin_amdgcn_wmma_f32_16x16x4_f32(false, a0, false, b0, (short)0, a00, false, false);
      a01 = __builtin_amdgcn_wmma_f32_16x16x4_f32(false, a0, false, b1, (short)0, a01, false, false);
      a10 = __builtin_amdgcn_wmma_f32_16x16x4_f32(false, a1, false, b0, (short)0, a10, false, false);
      a11 = __builtin_amdgcn_wmma_f32_16x16x4_f32(false, a1, false, b1, (short)0, a11, false, false);
    }
    __syncthreads();
  }

  // ---- dump U accumulators to LDS (sU aliases sQ, dead now) ----
  for (int v = 0; v < 8; ++v) {
    int r0 = rbase + v + 8 * half;      // D layout: lanes 0-15 -> M=v, lanes 16-31 -> M=v+8
    sU[r0][col]           = a00[v];
    sU[r0][col + 16]      = a01[v];
    sU[r0 + 16][col]      = a10[v];
    sU[r0 + 16][col + 16] = a11[v];
  }
  // inv_c is symmetric -> load untransposed, read rows for B fragments
  for (int idx = t; idx < 256; idx += 128) {
    int aa = idx >> 3, k4 = (idx & 7) << 2;
    *reinterpret_cast<float4*>(&sInv[aa][k4]) =
        *reinterpret_cast<const float4*>(&invP[(size_t)c * 1024 + (idx << 2)]);
  }
  __syncthreads();

  // ---- Phase 2: W = U(128x32) @ inv_c(32x32, zero-padded) ----
  v8f w00 = {}, w01 = {}, w10 = {}, w11 = {};
  for (int kk = 0; kk < 32; kk += 4) {
    v2f a0 = *reinterpret_cast<const v2f*>(&sU[mrow0][kk + 2 * half]);
    v2f a1 = *reinterpret_cast<const v2f*>(&sU[mrow1][kk + 2 * half]);
    v2f b0 = *reinterpret_cast<const v2f*>(&sInv[col][kk + 2 * half]);
    v2f b1 = *reinterpret_cast<const v2f*>(&sInv[col + 16][kk + 2 * half]);
    w00 = __builtin_amdgcn_wmma_f32_16x16x4_f32(false, a0, false, b0, (short)0, w00, false, false);
    w01 = __builtin_amdgcn_wmma_f32_16x16x4_f32(false, a0, false, b1, (short)0, w01, false, false);
    w10 = __builtin_amdgcn_wmma_f32_16x16x4_f32(false, a1, false, b0, (short)0, w10, false, false);
    w11 = __builtin_amdgcn_wmma_f32_16x16x4_f32(false, a1, false, b1, (short)0, w11, false, false);
  }
  for (int v = 0; v < 8; ++v) {
    int r0 = rbase + v + 8 * half;
    sW[r0][col]           = w00[v];
    sW[r0][col + 16]      = w01[v];
    sW[r0 + 16][col]      = w10[v];
    sW[r0 + 16][col + 16] = w11[v];
  }
  __syncthreads();

  // ---- Phase 3: per-row quadratic forms, accumulate into acc[b,c] ----
  const float rho = __expf(r[1]);
  const float lam = __expf(r[0]) * ((float)RESO / (float)DFEAT) + 1e-6f;
  const float c1  = rho * rho - 2.0f * rho;
  const float c2  = -rho * rho * lam;
  {
    float t1 = 0.0f, t2 = 0.0f;
    for (int j = 0; j < RESO; ++j) {
      float u = sU[t][j], w = sW[t][j];
      t1 += u * w;
      t2 += w * w;
    }
    int n = n0 + t;
    int b = n / RESO;
    atomicAdd(&accOut[b * NCLS + c], c1 * t1 + c2 * t2);
  }
}

// ---------------- finalize logits ----------------
__global__ void k_final(const float* __restrict__ qnorm, const float* __restrict__ acc,
                        const float* __restrict__ scale, float* __restrict__ out) {
  int idx = blockIdx.x * blockDim.x + threadIdx.x;
  if (idx >= BATCH * NCLS) return;
  int b = idx / NCLS;
  float qb = 0.0f;
  for (int s = 0; s < RESO; ++s) qb += qnorm[b * RESO + s];
  qb *= (1.0f / RESO);
  out[idx] = -scale[0] * (qb + acc[idx] * (1.0f / RESO));
}

extern "C" void kernel_launch(void* const* d_in, const int* in_sizes, int n_in,
                              void* d_out, int out_size, void* d_ws, size_t ws_size,
                              hipStream_t stream) {
  (void)in_sizes; (void)n_in; (void)out_size; (void)ws_size;
  const float* x     = (const float*)d_in[0];
  const float* proto = (const float*)d_in[1];
  const float* r     = (const float*)d_in[2];
  const float* scale = (const float*)d_in[3];
  float* out = (float*)d_out;

  float* ws    = (float*)d_ws;
  float* query = ws;                      // 3200*512
  float* qnorm = query + NQ * DFEAT;      // 3200
  float* invP  = qnorm + NQ;              // 100*1024 (zero-padded 32x32 per class)
  float* acc   = invP + NCLS * 1024;      // 128*100

  k_query<<<NQ, 128, 0, stream>>>(x, query, qnorm);
  k_inv<<<NCLS, 256, 0, stream>>>(proto, r, invP);
  k_zero<<<(BATCH * NCLS + 255) / 256, 256, 0, stream>>>(acc, BATCH * NCLS);
  dim3 g(NQ / 128, NCLS);
  k_main<<<g, 128, 0, stream>>>(query, proto, r, invP, acc);
  k_final<<<(BATCH * NCLS + 255) / 256, 256, 0, stream>>>(qnorm, acc, scale, out);
}